// VectorQuantizer_49357764165820
// MI455X (gfx1250) — compile-verified
//
#include <hip/hip_runtime.h>
#include <math.h>

// Problem dimensions (fixed by the reference).
#define VQ_B   32
#define VQ_D   256
#define VQ_HW  1024                  // 32*32
#define VQ_N   (VQ_B * VQ_HW)        // 32768 latent vectors
#define VQ_K   1024                  // codebook size
#define VQ_ELEMS (VQ_B * VQ_D * VQ_HW) // 8388608 elements of z_e / z_q_out

typedef float v2f __attribute__((ext_vector_type(2)));
typedef float v8f __attribute__((ext_vector_type(8)));

// Monotone float -> uint mapping: a < b  <=>  key(a) < key(b).
__device__ __forceinline__ unsigned int float_orderable(float f) {
  unsigned int u = __float_as_uint(f);
  return (u & 0x80000000u) ? ~u : (u | 0x80000000u);
}

// ---------------------------------------------------------------------------
// Workspace init: argmin keys = +inf, histogram = 0, loss accumulator = 0.
// ---------------------------------------------------------------------------
__global__ void vq_init_kernel(unsigned long long* __restrict__ keys,
                               unsigned int* __restrict__ counts,
                               float* __restrict__ loss_acc) {
  int n = blockIdx.x * blockDim.x + threadIdx.x;
  if (n < VQ_N)  keys[n] = ~0ull;
  if (n < VQ_K)  counts[n] = 0u;
  if (n == 0)    loss_acc[0] = 0.0f;
}

// ---------------------------------------------------------------------------
// ||e_k||^2 for each code row (K x D row-major).
// ---------------------------------------------------------------------------
__global__ void vq_enorm_kernel(const float* __restrict__ emb,
                                float* __restrict__ e_norm) {
  int k = blockIdx.x * blockDim.x + threadIdx.x;
  if (k >= VQ_K) return;
  const float* row = emb + ((size_t)k << 8);   // * VQ_D
  float s = 0.0f;
  #pragma unroll 8
  for (int d = 0; d < VQ_D; ++d) { float v = row[d]; s += v * v; }
  e_norm[k] = s;
}

// ---------------------------------------------------------------------------
// ||x_n||^2. z_e is (B, D, H, W); n = b*HW + hw. Consecutive threads read
// consecutive hw addresses at each d -> fully coalesced.
// ---------------------------------------------------------------------------
__global__ void vq_xnorm_kernel(const float* __restrict__ z_e,
                                float* __restrict__ x_norm) {
  int n = blockIdx.x * blockDim.x + threadIdx.x;
  if (n >= VQ_N) return;
  int b  = n >> 10;
  int hw = n & (VQ_HW - 1);
  const float* p = z_e + (((size_t)b * VQ_D) << 10) + hw;
  float s = 0.0f;
  #pragma unroll 4
  for (int d = 0; d < VQ_D; ++d) { float v = p[(size_t)d << 10]; s += v * v; }
  x_norm[n] = s;
}

// ---------------------------------------------------------------------------
// Distances GEMM (WMMA f32 16x16x4) + fused per-row argmin.
// Block = 256 threads (8 waves): 16 N-rows x 128 codes. Each wave owns one
// 16x16 tile and walks D in steps of 4 (64 WMMA ops).
// LDS A-tile uses a pair-interleaved layout so each A fragment is one b64 read:
//   elem(d, m) -> (d>>1)*32 + 2*m + (d&1)
// ---------------------------------------------------------------------------
__global__ __launch_bounds__(256) void vq_dist_kernel(
    const float* __restrict__ z_e, const float* __restrict__ emb,
    const float* __restrict__ x_norm, const float* __restrict__ e_norm,
    float* __restrict__ dist, unsigned long long* __restrict__ keys) {
  __shared__ float Atile[VQ_D * 16];           // 16 KB

  const int t        = threadIdx.x;
  const int n_base   = blockIdx.x << 4;        // 16 rows per block
  const int code_blk = blockIdx.y << 7;        // 128 codes per block
  const int batch    = n_base >> 10;           // 1024 rows per image => no straddle
  const int hw       = n_base & (VQ_HW - 1);

  // Cooperative A-tile load: flat[n_base + n_off][d] = z_e[batch, d, hw + n_off]
  {
    const int n_off = t & 15;
    const int d0    = t >> 4;
    const float* src = z_e + (((size_t)batch * VQ_D) << 10) + hw + n_off;
    #pragma unroll
    for (int p = 0; p < 16; ++p) {
      int dd = d0 + (p << 4);
      Atile[((dd >> 1) << 5) + (n_off << 1) + (dd & 1)] = src[(size_t)dd << 10];
    }
  }
  __syncthreads();

  const int wave  = t >> 5;
  const int lane  = t & 31;
  const int lhalf = lane >> 4;                 // 0: K=0,1   1: K=2,3
  const int l15   = lane & 15;
  const int kb    = lhalf << 1;
  const int code_base = code_blk + (wave << 4);

  const float* brow = emb + ((size_t)(code_base + l15) << 8);
  const float  en   = e_norm[code_base + l15];

  v8f acc = {};
  for (int d = 0; d < VQ_D; d += 4) {
    // A fragment: A[m = l15][k = kb, kb+1]  (ISA 16x4 layout)
    const float2 av = *(const float2*)&Atile[(d + kb) * 16 + (l15 << 1)];
    // B fragment: B[k = kb, kb+1][n = l15] = emb[code][d + kb .. +1]
    const float2 bv = *(const float2*)(brow + d + kb);
    v2f a, bm;
    a[0] = av.x; a[1] = av.y;
    bm[0] = bv.x; bm[1] = bv.y;
    acc = __builtin_amdgcn_wmma_f32_16x16x4_f32(
        /*neg_a=*/false, a, /*neg_b=*/false, bm,
        /*c_mod=*/(short)0, acc, /*reuse_a=*/false, /*reuse_b=*/false);
  }

  // Epilogue: dist = ||x||^2 + ||e||^2 - 2 x.e ; write + hierarchical argmin.
  #pragma unroll
  for (int v = 0; v < 8; ++v) {
    const int   row  = n_base + v + (lhalf << 3);       // C/D: M = v + 8*half
    const float dval = x_norm[row] + en - 2.0f * acc[v];
    dist[((size_t)row << 10) + code_base + l15] = dval;

    unsigned long long key =
        ((unsigned long long)float_orderable(dval) << 32) |
        (unsigned int)(code_base + l15);
    // min over the 16 lanes of this half (all share the same row)
    #pragma unroll
    for (int m = 8; m >= 1; m >>= 1) {
      unsigned long long o = __shfl_xor(key, m, 32);
      if (o < key) key = o;
    }
    if (l15 == 0) atomicMin(&keys[row], key);
  }
}

// ---------------------------------------------------------------------------
// Extract argmin indices from packed keys; build histogram.
// ---------------------------------------------------------------------------
__global__ void vq_index_kernel(const unsigned long long* __restrict__ keys,
                                int* __restrict__ idx_out,
                                unsigned int* __restrict__ counts) {
  int n = blockIdx.x * blockDim.x + threadIdx.x;
  if (n >= VQ_N) return;
  int idx = (int)(unsigned int)(keys[n] & 0xFFFFFFFFull);
  idx_out[n] = idx;
  atomicAdd(&counts[idx], 1u);
}

// ---------------------------------------------------------------------------
// z_q_out (B,D,H,W) = embedding[idx[n]][d] (straight-through == z_q),
// plus sum of squared differences for the loss (block-reduced, one atomic).
// ---------------------------------------------------------------------------
__global__ __launch_bounds__(256) void vq_gather_kernel(
    const float* __restrict__ z_e, const float* __restrict__ emb,
    const int* __restrict__ idx, float* __restrict__ zq_out,
    float* __restrict__ loss_acc) {
  size_t i = (size_t)blockIdx.x * 256 + threadIdx.x;  // flat (B,D,H,W) index
  size_t hw = i & (VQ_HW - 1);
  size_t bd = i >> 10;
  int d = (int)(bd & (VQ_D - 1));
  int b = (int)(bd >> 8);
  int n = (b << 10) + (int)hw;

  float e = emb[((size_t)idx[n] << 8) + d];
  float z = z_e[i];
  zq_out[i] = e;
  float diff = e - z;
  float v = diff * diff;

  const int lane = threadIdx.x & 31;
  const int wave = threadIdx.x >> 5;
  #pragma unroll
  for (int m = 16; m >= 1; m >>= 1) v += __shfl_xor(v, m, 32);
  __shared__ float wsum[8];
  if (lane == 0) wsum[wave] = v;
  __syncthreads();
  if (threadIdx.x < 8) {
    float s = wsum[threadIdx.x];
    #pragma unroll
    for (int m = 4; m >= 1; m >>= 1) s += __shfl_xor(s, m, 8);
    if (threadIdx.x == 0) atomicAdd(loss_acc, s);
  }
}

// ---------------------------------------------------------------------------
// Finalize: loss = 1.25 * mean((z_q - z)^2); perplexity from histogram.
// ---------------------------------------------------------------------------
__global__ __launch_bounds__(1024) void vq_final_kernel(
    const unsigned int* __restrict__ counts, const float* __restrict__ loss_acc,
    float* __restrict__ out_loss, float* __restrict__ out_perp) {
  int t = threadIdx.x;                                   // 0..1023 = code id
  float p = (float)counts[t] * (1.0f / (float)VQ_N);
  float v = p * logf(p + 1e-10f);
  #pragma unroll
  for (int m = 16; m >= 1; m >>= 1) v += __shfl_xor(v, m, 32);
  __shared__ float red[32];
  if ((t & 31) == 0) red[t >> 5] = v;
  __syncthreads();
  if (t < 32) {
    float s = red[t];
    #pragma unroll
    for (int m = 16; m >= 1; m >>= 1) s += __shfl_xor(s, m, 32);
    if (t == 0) {
      out_perp[0] = expf(-s);
      out_loss[0] = (1.0f + 0.25f) * loss_acc[0] *
                    (1.0f / ((float)VQ_N * (float)VQ_D));
    }
  }
}

// ---------------------------------------------------------------------------
// Host launcher.
// Output layout (floats, concatenated in reference return order):
//   [0 .. 8388607]                z_q_out (B,D,H,W)
//   [8388608]                     loss
//   [8388609]                     perplexity
//   [8388610 .. 8421377]          encoding_indices (int32 bits)
//   [8421378 .. 41975809]         distances (N,K)
// ---------------------------------------------------------------------------
extern "C" void kernel_launch(void* const* d_in, const int* in_sizes, int n_in,
                              void* d_out, int out_size, void* d_ws, size_t ws_size,
                              hipStream_t stream) {
  const float* z_e = (const float*)d_in[0];
  const float* emb = (const float*)d_in[1];

  float* out      = (float*)d_out;
  float* zq_out   = out;
  float* out_loss = out + (size_t)VQ_ELEMS;
  float* out_perp = out + (size_t)VQ_ELEMS + 1;
  int*   idx_out  = (int*)(out + (size_t)VQ_ELEMS + 2);
  float* dist_out = out + (size_t)VQ_ELEMS + 2 + VQ_N;

  // Workspace layout (bytes): x_norm[32768]f, e_norm[1024]f, keys[32768]u64,
  // counts[1024]u32, loss_acc[1]f  => ~392 KB total.
  char* ws = (char*)d_ws;
  float*              x_norm   = (float*)(ws);
  float*              e_norm   = (float*)(ws + 131072);
  unsigned long long* keys     = (unsigned long long*)(ws + 135168);
  unsigned int*       counts   = (unsigned int*)(ws + 397312);
  float*              loss_acc = (float*)(ws + 401408);

  vq_init_kernel <<<VQ_N / 256, 256, 0, stream>>>(keys, counts, loss_acc);
  vq_enorm_kernel<<<VQ_K / 256, 256, 0, stream>>>(emb, e_norm);
  vq_xnorm_kernel<<<VQ_N / 256, 256, 0, stream>>>(z_e, x_norm);

  dim3 gdist(VQ_N / 16, VQ_K / 128);
  vq_dist_kernel<<<gdist, 256, 0, stream>>>(z_e, emb, x_norm, e_norm,
                                            dist_out, keys);

  vq_index_kernel<<<VQ_N / 256, 256, 0, stream>>>(keys, idx_out, counts);
  vq_gather_kernel<<<VQ_ELEMS / 256, 256, 0, stream>>>(z_e, emb, idx_out,
                                                       zq_out, loss_acc);
  vq_final_kernel<<<1, 1024, 0, stream>>>(counts, loss_acc, out_loss, out_perp);
}